// TransE_33079838114366
// MI455X (gfx1250) — compile-verified
//
#include <hip/hip_runtime.h>
#include <hip/hip_bf16.h>
#include <stdint.h>

#define NUM_ENT   100000
#define EMB       128
#define BATCH     64
#define GAMMA_F   12.0f

#define TILE_E    32          // entities per block (lane-mapped); 100000/32 = 3125 exactly
#define PAD_ROW   132         // dwords per LDS row (128 + 4 pad, keeps 16B alignment, kills bank conflicts)
#define BG        8           // batches per wave (8 waves * 8 batches = 64)

// ---------------------------------------------------------------------------
// Kernel 1: q[b][d] = ent_w[sub[b]][d] + rel_w[rel[b]][d]   (64 x 128 -> d_ws)
// ---------------------------------------------------------------------------
__global__ __launch_bounds__(EMB)
void transe_build_query(const int* __restrict__ sub, const int* __restrict__ rel,
                        const float* __restrict__ ent_w, const float* __restrict__ rel_w,
                        float* __restrict__ q) {
    const int b = blockIdx.x;     // 0..63
    const int d = threadIdx.x;    // 0..127
    const int s = sub[b];
    const int r = rel[b];
    q[b * EMB + d] = ent_w[s * EMB + d] + rel_w[r * EMB + d];
}

// ---------------------------------------------------------------------------
// Kernel 2: dist + sigmoid. Block = 256 threads (8 waves), 32 entities/block.
//   lane  -> entity within tile
//   wave  -> batch group of 8 (uniform per wave => q loads are scalar SMEM)
// Entity tile staged global->LDS with the CDNA5 async-tensor path.
// ---------------------------------------------------------------------------
__global__ __launch_bounds__(256, 4)
void transe_scores(const float* __restrict__ ent_w, const float* __restrict__ q,
                   float* __restrict__ out) {
    __shared__ __align__(16) float wlds[TILE_E * PAD_ROW];   // 16896 B

    const int tid  = threadIdx.x;
    const int e0   = blockIdx.x * TILE_E;
    const int lane = tid & 31;
    const int wave = tid >> 5;

    // ---- Async stage: 32 rows x 128 f32 = 1024 x b128, 4 per thread --------
    // LDS offset of a generic __shared__ pointer is its low 32 bits.
    const uint32_t lds_base = (uint32_t)(uintptr_t)(&wlds[0]);
#pragma unroll
    for (int i = 0; i < 4; ++i) {
        const int idx = i * 256 + tid;         // 0..1023
        const int row = idx >> 5;              // 0..31  (32 float4 per row)
        const int c4  = idx & 31;              // float4 column
        const float*   src    = ent_w + (size_t)(e0 + row) * EMB + c4 * 4;
        const uint32_t ldsoff = lds_base + (uint32_t)(row * PAD_ROW + c4 * 4) * 4u;
        asm volatile("global_load_async_to_lds_b128 %0, %1, off"
                     :: "v"(ldsoff), "v"(src) : "memory");
    }
    asm volatile("s_wait_asynccnt 0x0" ::: "memory");
    __syncthreads();

    // ---- Compute: acc[j] = sum_d |q[bg*8+j][d] - w[e][d]| ------------------
    // Force the q base index wave-uniform so the backend uses SMEM loads.
    const int bg = __builtin_amdgcn_readfirstlane(wave);
    const float* __restrict__ qbase = q + (size_t)bg * BG * EMB;
    __builtin_prefetch(qbase, 0, 0);   // global_prefetch_b8 (q is tiny & hot)

    float acc[BG];
#pragma unroll
    for (int j = 0; j < BG; ++j) acc[j] = 0.0f;

    const float* __restrict__ wrow = &wlds[lane * PAD_ROW];

#pragma unroll 2
    for (int c = 0; c < EMB / 4; ++c) {                    // 32 chunks of 4 dims
        const float4 wv = *reinterpret_cast<const float4*>(wrow + c * 4);
#pragma unroll
        for (int j = 0; j < BG; ++j) {
            const float4 qv = *reinterpret_cast<const float4*>(qbase + j * EMB + c * 4);
            acc[j] += fabsf(qv.x - wv.x);                  // v_add_f32 with |src| modifier
            acc[j] += fabsf(qv.y - wv.y);
            acc[j] += fabsf(qv.z - wv.z);
            acc[j] += fabsf(qv.w - wv.w);
        }
    }

    // ---- Epilogue: sigmoid(GAMMA - dist), coalesced non-temporal stores ----
    float* __restrict__ orow = out + (size_t)(wave * BG) * NUM_ENT + e0 + lane;
#pragma unroll
    for (int j = 0; j < BG; ++j) {
        const float s = 1.0f / (1.0f + __expf(acc[j] - GAMMA_F));
        __builtin_nontemporal_store(s, orow + (size_t)j * NUM_ENT);
    }
}

// ---------------------------------------------------------------------------
extern "C" void kernel_launch(void* const* d_in, const int* in_sizes, int n_in,
                              void* d_out, int out_size, void* d_ws, size_t ws_size,
                              hipStream_t stream) {
    const int*   sub   = (const int*)  d_in[0];
    const int*   rel   = (const int*)  d_in[1];
    const float* ent_w = (const float*)d_in[2];
    const float* rel_w = (const float*)d_in[3];
    float*       out   = (float*)d_out;
    float*       q     = (float*)d_ws;          // 64*128*4 = 32 KB scratch

    transe_build_query<<<BATCH, EMB, 0, stream>>>(sub, rel, ent_w, rel_w, q);
    transe_scores<<<NUM_ENT / TILE_E, 256, 0, stream>>>(ent_w, q, out);
}